// LNN6DOF_60662118088750
// MI455X (gfx1250) — compile-verified
//
#include <hip/hip_runtime.h>
#include <math.h>

typedef float v2f __attribute__((ext_vector_type(2)));
typedef float v8f __attribute__((ext_vector_type(8)));

namespace {

constexpr int AS  = 132;   // padded activation / W2,W3 row stride (floats) -> LDS bank-conflict free
constexpr int W1S = 14;    // padded W1^T row stride
constexpr float DTc   = 0.02f;
constexpr float ALPHA = 2.0f / 3.0f;

constexpr int SZ_W1  = 128 * W1S;
constexpr int SZ_W2  = 128 * AS;
constexpr int SZ_W3L = 32 * AS;    // 21 cols padded to 32
constexpr int SZ_W3V = 16 * AS;    // 1 col padded to 16

constexpr int O_W1L = 0;
constexpr int O_W2L = O_W1L + SZ_W1;
constexpr int O_W3L = O_W2L + SZ_W2;
constexpr int O_W1V = O_W3L + SZ_W3L;
constexpr int O_W2V = O_W1V + SZ_W1;
constexpr int O_W3V = O_W2V + SZ_W2;
constexpr int O_B1L = O_W3V + SZ_W3V;
constexpr int O_B2L = O_B1L + 128;
constexpr int O_B3L = O_B2L + 128;
constexpr int O_B1V = O_B3L + 32;
constexpr int O_B2V = O_B1V + 128;
constexpr int O_B3V = O_B2V + 128;
constexpr int O_WAVE = O_B3V + 16;

constexpr int WV_ACT  = 0;
constexpr int WV_SIG1 = 16 * AS;
constexpr int WV_SIG2 = 2 * 16 * AS;
constexpr int WV_Y    = 3 * 16 * AS;
constexpr int WV_DY   = WV_Y + 16 * 21;
constexpr int WV_DV   = WV_DY + 16 * 126;
constexpr int WAVE_FLOATS = WV_DV + 16 * 6;

constexpr int SMEM_FLOATS = O_WAVE + 4 * WAVE_FLOATS;   // ~79408 floats = ~310 KB

enum { MODE_LIN = 0, MODE_SOFT = 1, MODE_TANG = 2 };

struct SmemPtrs {
  float *W1Lt, *W2Lt, *W3Lt, *W1Vt, *W2Vt, *W3Vt;
  float *b1L, *b2L, *b3L, *b1V, *b2V, *b3V;
  float *act, *sig1, *sig2, *y, *dy, *dV;
};

// out(16 x ntiles*16) = act(16 x K) @ Wt^T  (Wt stored transposed: [outcol][inrow])
// MODE_SOFT: z+=bias; store softplus(z) to dst and sigmoid(z) to sig
// MODE_TANG: dst = sig * z   (JVP through softplus, no bias)
// MODE_LIN : dst = z (+bias if non-null)
template <int NT, int MODE>
__device__ __forceinline__ void mlp_layer(const float* actb, const float* Wt, int wstride, int K,
                                          const float* bias, float* sig,
                                          float* dst, int dstride, int ncols, int lane) {
  const int m  = lane & 15;
  const int hi = lane >> 4;
  const v8f vzero = {0.f, 0.f, 0.f, 0.f, 0.f, 0.f, 0.f, 0.f};
  v8f acc[NT];
#pragma unroll
  for (int t = 0; t < NT; ++t) acc[t] = vzero;

  for (int k = 0; k < K; k += 4) {
    const int kc = k + 2 * hi;
    v2f av;
    av.x = actb[m * AS + kc];
    av.y = actb[m * AS + kc + 1];
#pragma unroll
    for (int t = 0; t < NT; ++t) {
      const int wr = t * 16 + m;          // output column owned by this lane
      v2f bv;
      bv.x = Wt[wr * wstride + kc];
      bv.y = Wt[wr * wstride + kc + 1];
      acc[t] = __builtin_amdgcn_wmma_f32_16x16x4_f32(false, av, false, bv,
                                                     (short)0, acc[t], false, false);
    }
  }

#pragma unroll
  for (int t = 0; t < NT; ++t) {
#pragma unroll
    for (int r = 0; r < 8; ++r) {
      const int mr  = r + 8 * hi;         // batch row
      const int col = t * 16 + m;         // output column
      if (col < ncols) {
        float z = acc[t][r];
        if (MODE == MODE_SOFT) {
          z += bias[col];
          sig[mr * AS + col] = 1.f / (1.f + expf(-z));
          dst[mr * dstride + col] = fmaxf(z, 0.f) + log1pf(expf(-fabsf(z)));
        } else if (MODE == MODE_TANG) {
          dst[mr * dstride + col] = sig[mr * AS + col] * z;
        } else {
          dst[mr * dstride + col] = bias ? (z + bias[col]) : z;
        }
      }
    }
  }
}

__device__ __forceinline__ void load_wt_T(float* dst, const float* src, int fin, int fout,
                                          int foutPad, int stride, int tid, int nthr) {
  const int total = foutPad * fin;
  for (int p = tid; p < total; p += nthr) {
    const int o = p / fin;
    const int i = p - o * fin;
    dst[o * stride + i] = (o < fout) ? src[i * fout + o] : 0.f;
  }
}

__device__ __forceinline__ void load_b(float* dst, const float* src, int n, int npad,
                                       int tid, int nthr) {
  for (int p = tid; p < npad; p += nthr) dst[p] = (p < n) ? src[p] : 0.f;
}

// One derivs() evaluation for 16 batch elements owned by this wave.
__device__ __forceinline__ void derivs_eval(const SmemPtrs& S, int lane,
                                            const float* q, const float* qd, const float* aI,
                                            float* qdd) {
  const int m = lane & 15;
  const bool owner = lane < 16;
  float* act = S.act;

  // ---------- mass net: primal ----------
  if (owner) {
#pragma unroll
    for (int i = 0; i < 6; ++i) {
      act[m * AS + 2 * i]     = cosf(q[i]);
      act[m * AS + 2 * i + 1] = sinf(q[i]);
    }
  }
  mlp_layer<8, MODE_SOFT>(act, S.W1Lt, W1S, 12, S.b1L, S.sig1, act, AS, 128, lane);
  mlp_layer<8, MODE_SOFT>(act, S.W2Lt, AS, 128, S.b2L, S.sig2, act, AS, 128, lane);
  mlp_layer<2, MODE_LIN >(act, S.W3Lt, AS, 128, S.b3L, nullptr, S.y, 21, 21, lane);

  // ---------- mass net: 6 forward tangents (jacfwd) ----------
  for (int d = 0; d < 6; ++d) {
    if (owner) {
#pragma unroll
      for (int c2 = 0; c2 < 12; ++c2) act[m * AS + c2] = 0.f;
      act[m * AS + 2 * d]     = -sinf(q[d]);
      act[m * AS + 2 * d + 1] =  cosf(q[d]);
    }
    mlp_layer<8, MODE_TANG>(act, S.W1Lt, W1S, 12, nullptr, S.sig1, act, AS, 128, lane);
    mlp_layer<8, MODE_TANG>(act, S.W2Lt, AS, 128, nullptr, S.sig2, act, AS, 128, lane);
    mlp_layer<2, MODE_LIN >(act, S.W3Lt, AS, 128, nullptr, nullptr, S.dy + d * 21, 126, 21, lane);
  }

  // ---------- potential net: primal (to get sigmas) ----------
  if (owner) {
#pragma unroll
    for (int i = 0; i < 6; ++i) {
      act[m * AS + 2 * i]     = cosf(q[i]);
      act[m * AS + 2 * i + 1] = sinf(q[i]);
    }
  }
  mlp_layer<8, MODE_SOFT>(act, S.W1Vt, W1S, 12, S.b1V, S.sig1, act, AS, 128, lane);
  mlp_layer<8, MODE_SOFT>(act, S.W2Vt, AS, 128, S.b2V, S.sig2, act, AS, 128, lane);

  // ---------- potential net: 6 tangents -> dV/dq ----------
  for (int d = 0; d < 6; ++d) {
    if (owner) {
#pragma unroll
      for (int c2 = 0; c2 < 12; ++c2) act[m * AS + c2] = 0.f;
      act[m * AS + 2 * d]     = -sinf(q[d]);
      act[m * AS + 2 * d + 1] =  cosf(q[d]);
    }
    mlp_layer<8, MODE_TANG>(act, S.W1Vt, W1S, 12, nullptr, S.sig1, act, AS, 128, lane);
    mlp_layer<8, MODE_TANG>(act, S.W2Vt, AS, 128, nullptr, S.sig2, act, AS, 128, lane);
    mlp_layer<1, MODE_LIN >(act, S.W3Vt, AS, 128, nullptr, nullptr, S.dV + d, 6, 1, lane);
  }

  // ---------- per-element 6x6 linear algebra (owner lanes) ----------
  if (owner) {
    const float* ye  = S.y  + m * 21;
    const float* dye = S.dy + m * 126;
    const float* dVe = S.dV + m * 6;

    float Lm[6][6], Mm[6][6];
#pragma unroll
    for (int r = 0; r < 6; ++r)
#pragma unroll
      for (int c = 0; c < 6; ++c) Lm[r][c] = 0.f;
#pragma unroll
    for (int r = 0; r < 6; ++r)
#pragma unroll
      for (int c = 0; c < 6; ++c)
        if (c <= r) Lm[r][c] = ye[r * (r + 1) / 2 + c];

#pragma unroll
    for (int r = 0; r < 6; ++r)
#pragma unroll
      for (int c = 0; c < 6; ++c) {
        float s = 0.f;
#pragma unroll
        for (int k = 0; k < 6; ++k) s += Lm[r][k] * Lm[c][k];
        Mm[r][c] = s + ((r == c) ? 1e-6f : 0.f);
      }

    float cv[6] = {0.f, 0.f, 0.f, 0.f, 0.f, 0.f};
#pragma unroll
    for (int d = 0; d < 6; ++d) {
      float dL[6][6];
#pragma unroll
      for (int r = 0; r < 6; ++r)
#pragma unroll
        for (int c = 0; c < 6; ++c)
          dL[r][c] = (c <= r) ? dye[d * 21 + r * (r + 1) / 2 + c] : 0.f;

      float G[6][6];
#pragma unroll
      for (int j = 0; j < 6; ++j)
#pragma unroll
        for (int k = 0; k < 6; ++k) {
          float s = 0.f;
#pragma unroll
          for (int t = 0; t < 6; ++t) s += dL[j][t] * Lm[k][t] + Lm[j][t] * dL[k][t];
          G[j][k] = s;
        }

      float P[6][6];   // P = G * M  (M symmetric)
#pragma unroll
      for (int j = 0; j < 6; ++j)
#pragma unroll
        for (int l = 0; l < 6; ++l) {
          float s = 0.f;
#pragma unroll
          for (int k = 0; k < 6; ++k) s += G[j][k] * Mm[l][k];
          P[j][l] = s;
        }

      float v[6];      // v = (P + P^T) qd = H_d qd
#pragma unroll
      for (int i = 0; i < 6; ++i) {
        float s = 0.f;
#pragma unroll
        for (int l = 0; l < 6; ++l) s += (P[i][l] + P[l][i]) * qd[l];
        v[i] = s;
      }
#pragma unroll
      for (int i = 0; i < 6; ++i) cv[i] += qd[d] * v[i];
      float s2 = 0.f;
#pragma unroll
      for (int i = 0; i < 6; ++i) s2 += qd[i] * v[i];
      cv[d] -= 0.5f * s2;
    }

    // inverse of tril(M) by forward substitution
    float Li[6][6];
#pragma unroll
    for (int r = 0; r < 6; ++r)
#pragma unroll
      for (int c = 0; c < 6; ++c) Li[r][c] = 0.f;
#pragma unroll
    for (int i = 0; i < 6; ++i) Li[i][i] = 1.f / Mm[i][i];
#pragma unroll
    for (int i = 1; i < 6; ++i)
#pragma unroll
      for (int j = 0; j < i; ++j) {
        float s = 0.f;
#pragma unroll
        for (int k = 0; k < 6; ++k)
          if (k >= j && k < i) s += Mm[i][k] * Li[k][j];
        Li[i][j] = -s * Li[i][i];
      }

    float Mi[6][6];  // Minv = Li^T Li
#pragma unroll
    for (int r = 0; r < 6; ++r)
#pragma unroll
      for (int c = 0; c < 6; ++c) {
        float s = 0.f;
#pragma unroll
        for (int k = 0; k < 6; ++k) s += Li[k][r] * Li[k][c];
        Mi[r][c] = s;
      }

    float rhs[6];
#pragma unroll
    for (int i = 0; i < 6; ++i) rhs[i] = aI[i] - cv[i] - dVe[i];
#pragma unroll
    for (int i = 0; i < 6; ++i) {
      float s = 0.f;
#pragma unroll
      for (int j = 0; j < 6; ++j) s += Mi[i][j] * rhs[j];
      qdd[i] = s;
    }
  }
}

__global__ __launch_bounds__(128) void lnn6_rk2_wmma(
    const float* __restrict__ o, const float* __restrict__ aIn,
    const float* __restrict__ gW1L, const float* __restrict__ gb1L,
    const float* __restrict__ gW2L, const float* __restrict__ gb2L,
    const float* __restrict__ gW3L, const float* __restrict__ gb3L,
    const float* __restrict__ gW1V, const float* __restrict__ gb1V,
    const float* __restrict__ gW2V, const float* __restrict__ gb2V,
    const float* __restrict__ gW3V, const float* __restrict__ gb3V,
    float* __restrict__ outp, int nElem) {
  extern __shared__ float smem[];
  const int tid = threadIdx.x;
  const int nthr = 128;

  // cooperative load of all weights into LDS, transposed + padded
  load_wt_T(smem + O_W1L, gW1L, 12, 128, 128, W1S, tid, nthr);
  load_wt_T(smem + O_W2L, gW2L, 128, 128, 128, AS, tid, nthr);
  load_wt_T(smem + O_W3L, gW3L, 128, 21, 32, AS, tid, nthr);
  load_wt_T(smem + O_W1V, gW1V, 12, 128, 128, W1S, tid, nthr);
  load_wt_T(smem + O_W2V, gW2V, 128, 128, 128, AS, tid, nthr);
  load_wt_T(smem + O_W3V, gW3V, 128, 1, 16, AS, tid, nthr);
  load_b(smem + O_B1L, gb1L, 128, 128, tid, nthr);
  load_b(smem + O_B2L, gb2L, 128, 128, tid, nthr);
  load_b(smem + O_B3L, gb3L, 21, 32, tid, nthr);
  load_b(smem + O_B1V, gb1V, 128, 128, tid, nthr);
  load_b(smem + O_B2V, gb2V, 128, 128, tid, nthr);
  load_b(smem + O_B3V, gb3V, 1, 16, tid, nthr);
  __syncthreads();

  const int wave = tid >> 5;
  const int lane = tid & 31;
  const int m = lane & 15;

  SmemPtrs S;
  S.W1Lt = smem + O_W1L; S.W2Lt = smem + O_W2L; S.W3Lt = smem + O_W3L;
  S.W1Vt = smem + O_W1V; S.W2Vt = smem + O_W2V; S.W3Vt = smem + O_W3V;
  S.b1L = smem + O_B1L; S.b2L = smem + O_B2L; S.b3L = smem + O_B3L;
  S.b1V = smem + O_B1V; S.b2V = smem + O_B2V; S.b3V = smem + O_B3V;
  float* wbase = smem + O_WAVE + wave * WAVE_FLOATS;
  S.act = wbase + WV_ACT; S.sig1 = wbase + WV_SIG1; S.sig2 = wbase + WV_SIG2;
  S.y = wbase + WV_Y; S.dy = wbase + WV_DY; S.dV = wbase + WV_DV;

  const int idx = blockIdx.x * 64 + wave * 16 + m;
  const bool valid = (lane < 16) && (idx < nElem);

  float q[6]  = {0.f, 0.f, 0.f, 0.f, 0.f, 0.f};
  float qv[6] = {0.f, 0.f, 0.f, 0.f, 0.f, 0.f};
  float av[6] = {0.f, 0.f, 0.f, 0.f, 0.f, 0.f};
  if (valid) {
#pragma unroll
    for (int i = 0; i < 6; ++i) {
      q[i]  = o[idx * 18 + i];
      qv[i] = o[idx * 18 + 6 + i];
      av[i] = aIn[idx * 6 + i];
    }
  }

  float qdd1[6] = {0.f, 0.f, 0.f, 0.f, 0.f, 0.f};
  derivs_eval(S, lane, q, qv, av, qdd1);

  float qm[6], qvm[6];
#pragma unroll
  for (int i = 0; i < 6; ++i) {
    qm[i]  = q[i]  + ALPHA * DTc * qv[i];
    qvm[i] = qv[i] + ALPHA * DTc * qdd1[i];
  }

  float qdd2[6] = {0.f, 0.f, 0.f, 0.f, 0.f, 0.f};
  derivs_eval(S, lane, qm, qvm, av, qdd2);

  if (valid) {
#pragma unroll
    for (int i = 0; i < 6; ++i)
      outp[idx * 18 + i] = q[i] + DTc * (0.25f * qv[i] + 0.75f * qvm[i]);
#pragma unroll
    for (int i = 0; i < 6; ++i)
      outp[idx * 18 + 6 + i] = qv[i] + DTc * (0.25f * qdd1[i] + 0.75f * qdd2[i]);
#pragma unroll
    for (int i = 0; i < 6; ++i)
      outp[idx * 18 + 12 + i] = o[idx * 18 + 12 + i];
  }
}

} // namespace

extern "C" void kernel_launch(void* const* d_in, const int* in_sizes, int n_in,
                              void* d_out, int out_size, void* d_ws, size_t ws_size,
                              hipStream_t stream) {
  (void)n_in; (void)out_size; (void)d_ws; (void)ws_size;
  const float* o   = (const float*)d_in[0];
  const float* a   = (const float*)d_in[1];
  const float* W1L = (const float*)d_in[2];
  const float* b1L = (const float*)d_in[3];
  const float* W2L = (const float*)d_in[4];
  const float* b2L = (const float*)d_in[5];
  const float* W3L = (const float*)d_in[6];
  const float* b3L = (const float*)d_in[7];
  const float* W1V = (const float*)d_in[8];
  const float* b1V = (const float*)d_in[9];
  const float* W2V = (const float*)d_in[10];
  const float* b2V = (const float*)d_in[11];
  const float* W3V = (const float*)d_in[12];
  const float* b3V = (const float*)d_in[13];

  const int nElem = in_sizes[0] / 18;               // 32768
  const int blocks = (nElem + 63) / 64;             // 64 elements per 128-thread block
  const size_t smem = (size_t)SMEM_FLOATS * sizeof(float);

  lnn6_rk2_wmma<<<blocks, 128, smem, stream>>>(
      o, a, W1L, b1L, W2L, b2L, W3L, b3L, W1V, b1V, W2V, b2V, W3V, b3V,
      (float*)d_out, nElem);
}